// GraphNet_1451698946377
// MI455X (gfx1250) — compile-verified
//
#include <hip/hip_runtime.h>
#include <math.h>
#include <stdint.h>

#define IN_DIM 128
#define HID    64
#define ATT    64
#define NEG_SLOPE 0.01f

typedef __attribute__((ext_vector_type(2))) float v2f;
typedef __attribute__((ext_vector_type(8))) float v8f;

// ---------------------------------------------------------------------------
// Kernel 0: fold attention matrices to vectors: wl[h] = sum_a W_l[h][a]
// ---------------------------------------------------------------------------
__global__ void gat_wsum(const float* __restrict__ Wl, const float* __restrict__ Wr,
                         float* __restrict__ wl, float* __restrict__ wr) {
    int h = threadIdx.x;            // 64 threads
    float s1 = 0.f, s2 = 0.f;
    for (int a = 0; a < ATT; ++a) {
        s1 += Wl[h * ATT + a];
        s2 += Wr[h * ATT + a];
    }
    wl[h] = s1;
    wr[h] = s2;
}

// ---------------------------------------------------------------------------
// Kernel 1: CSR row offsets from sorted edge_dst (lower_bound per node)
// ---------------------------------------------------------------------------
__global__ void gat_rowstart(const int* __restrict__ dst, int* __restrict__ rs,
                             int n, int e) {
    int t = blockIdx.x * blockDim.x + threadIdx.x;
    if (t > n) return;
    int lo = 0, hi = e;
    while (lo < hi) {
        int mid = (lo + hi) >> 1;
        if (dst[mid] < t) lo = mid + 1; else hi = mid;
    }
    rs[t] = lo;
}

// ---------------------------------------------------------------------------
// Kernel 2: WMMA fp32 GEMMs: ft = feats @ W_fc, base(out) = feats @ W_res
// Weights staged to LDS via CDNA5 async-to-LDS (ASYNCcnt), then one wave per
// 16-row tile runs 32 K-steps of v_wmma_f32_16x16x4_f32 over 8 output tiles.
// A-frag (16x4 f32): lanes 0-15 hold K=k0+{0,1}; lanes 16-31 hold K=k0+{2,3}.
// B-frag (4x16 f32): VGPR j = row K=k0+j+2*half, cols striped over 16 lanes.
// ---------------------------------------------------------------------------
__global__ __launch_bounds__(256)
void gat_gemm_wmma(const float* __restrict__ feats,
                   const float* __restrict__ Wfc,
                   const float* __restrict__ Wres,
                   float* __restrict__ ft,
                   float* __restrict__ base,
                   int n) {
    __shared__ float sWfc[IN_DIM * HID];
    __shared__ float sWres[IN_DIM * HID];

    int t = threadIdx.x;
    // Async global->LDS staging: 2 x 32KB, 16B per lane per issue.
    // LDS operand = low 32 bits of the flat pointer (= AS3 byte offset).
    for (int i = t * 4; i < IN_DIM * HID; i += 256 * 4) {
        uint32_t l1 = (uint32_t)(uintptr_t)&sWfc[i];
        uint64_t g1 = (uint64_t)(uintptr_t)&Wfc[i];
        asm volatile("global_load_async_to_lds_b128 %0, %1, off"
                     :: "v"(l1), "v"(g1) : "memory");
        uint32_t l2 = (uint32_t)(uintptr_t)&sWres[i];
        uint64_t g2 = (uint64_t)(uintptr_t)&Wres[i];
        asm volatile("global_load_async_to_lds_b128 %0, %1, off"
                     :: "v"(l2), "v"(g2) : "memory");
    }
    asm volatile("s_wait_asynccnt 0x0" ::: "memory");
    __syncthreads();

    int wave = t >> 5;
    int lane = t & 31;
    long tile = (long)blockIdx.x * 8 + wave;
    long row0 = tile * 16;
    if (row0 >= n) return;                 // wave-uniform: EXEC stays all-ones

    int half = lane >> 4;                  // 0 = lanes 0-15, 1 = lanes 16-31
    int lm   = lane & 15;

    v8f zero = {0.f, 0.f, 0.f, 0.f, 0.f, 0.f, 0.f, 0.f};
    v8f cfc[4]  = {zero, zero, zero, zero};
    v8f cres[4] = {zero, zero, zero, zero};

    long arow = row0 + lm;
    if (arow >= n) arow = n - 1;           // clamp, keep EXEC full
    const float* aptr = feats + arow * IN_DIM;

    for (int k0 = 0; k0 < IN_DIM; k0 += 4) {
        float2 av = *(const float2*)&aptr[k0 + half * 2];
        v2f a; a.x = av.x; a.y = av.y;
        int kb = k0 + 2 * half;
        #pragma unroll
        for (int nt = 0; nt < 4; ++nt) {
            int col = nt * 16 + lm;
            v2f bfc;
            bfc.x = sWfc[(kb + 0) * HID + col];
            bfc.y = sWfc[(kb + 1) * HID + col];
            cfc[nt] = __builtin_amdgcn_wmma_f32_16x16x4_f32(
                false, a, false, bfc, (short)0, cfc[nt], false, false);
            v2f brs;
            brs.x = sWres[(kb + 0) * HID + col];
            brs.y = sWres[(kb + 1) * HID + col];
            cres[nt] = __builtin_amdgcn_wmma_f32_16x16x4_f32(
                false, a, false, brs, (short)0, cres[nt], false, false);
        }
    }

    // C/D layout: VGPR r -> row M=r (lanes 0-15) / M=r+8 (lanes 16-31), N=lane%16
    if (row0 + 16 <= n) {
        // fast path: wave-uniform in-bounds, unguarded coalesced stores
        float* __restrict__ pft = ft   + (row0 + half * 8) * HID + lm;
        float* __restrict__ pbs = base + (row0 + half * 8) * HID + lm;
        #pragma unroll
        for (int nt = 0; nt < 4; ++nt) {
            #pragma unroll
            for (int r = 0; r < 8; ++r) {
                pft[r * HID + nt * 16] = cfc[nt][r];
                pbs[r * HID + nt * 16] = cres[nt][r];
            }
        }
    } else {
        #pragma unroll
        for (int nt = 0; nt < 4; ++nt) {
            int col = nt * 16 + lm;
            #pragma unroll
            for (int r = 0; r < 8; ++r) {
                long row = row0 + r + half * 8;
                if (row < n) {
                    ft[row * HID + col]   = cfc[nt][r];
                    base[row * HID + col] = cres[nt][r];
                }
            }
        }
    }
}

// ---------------------------------------------------------------------------
// Kernel 3: per-node attention scores: s_dst = ft . wl, s_src = ft . wr
// One wave32 per node, float2 per lane, shfl_xor reduction.
// ---------------------------------------------------------------------------
__global__ __launch_bounds__(256)
void gat_scores(const float* __restrict__ ft, const float* __restrict__ wl,
                const float* __restrict__ wr, float* __restrict__ sd,
                float* __restrict__ ss, int n) {
    int wave = threadIdx.x >> 5, lane = threadIdx.x & 31;
    long node = (long)blockIdx.x * 8 + wave;
    if (node >= n) return;
    float2 f = *(const float2*)&ft[node * HID + lane * 2];
    float2 a = *(const float2*)&wl[lane * 2];
    float2 b = *(const float2*)&wr[lane * 2];
    float s1 = f.x * a.x + f.y * a.y;
    float s2 = f.x * b.x + f.y * b.y;
    for (int off = 16; off; off >>= 1) {
        s1 += __shfl_xor(s1, off, 32);
        s2 += __shfl_xor(s2, off, 32);
    }
    if (lane == 0) { sd[node] = s1; ss[node] = s2; }
}

// ---------------------------------------------------------------------------
// Kernel 4: segment softmax + weighted aggregation + residual + ELU.
// One wave32 per dst node; deterministic (no atomics, sorted CSR ranges).
// ---------------------------------------------------------------------------
__global__ __launch_bounds__(256)
void gat_aggregate(const float* __restrict__ ft, const float* __restrict__ sdv,
                   const float* __restrict__ ss, const int* __restrict__ esrc,
                   const int* __restrict__ rs, float* __restrict__ out, int n) {
    int wave = threadIdx.x >> 5, lane = threadIdx.x & 31;
    long node = (long)blockIdx.x * 8 + wave;
    if (node >= n) return;

    int start = rs[node], end = rs[node + 1];
    float sd = sdv[node];

    // pass 1: lane-parallel running max of leaky_relu scores
    float m = -3.402823466e38f;
    for (int e = start + lane; e < end; e += 32) {
        float sc = sd + ss[esrc[e]];
        sc = sc > 0.f ? sc : NEG_SLOPE * sc;
        m = fmaxf(m, sc);
    }
    for (int off = 16; off; off >>= 1) m = fmaxf(m, __shfl_xor(m, off, 32));

    // pass 2: serial edges, feature-parallel accumulation (float2/lane)
    float denom = 0.f, ax = 0.f, ay = 0.f;
    for (int e = start; e < end; ++e) {
        int src = esrc[e];
        if (e + 1 < end)
            __builtin_prefetch(&ft[(long)esrc[e + 1] * HID + lane * 2], 0, 1);
        float sc = sd + ss[src];
        sc = sc > 0.f ? sc : NEG_SLOPE * sc;
        float w = __expf(sc - m);
        denom += w;
        float2 f = *(const float2*)&ft[(long)src * HID + lane * 2];
        ax += w * f.x;
        ay += w * f.y;
    }
    if (denom == 0.f) denom = 1.f;
    float inv = 1.f / denom;

    float2 b = *(const float2*)&out[node * HID + lane * 2];
    float rx = b.x + ax * inv;
    float ry = b.y + ay * inv;
    rx = rx > 0.f ? rx : __expf(rx) - 1.f;      // ELU(alpha=1)
    ry = ry > 0.f ? ry : __expf(ry) - 1.f;
    float2 r; r.x = rx; r.y = ry;
    *(float2*)&out[node * HID + lane * 2] = r;
}

// ---------------------------------------------------------------------------
extern "C" void kernel_launch(void* const* d_in, const int* in_sizes, int n_in,
                              void* d_out, int out_size, void* d_ws, size_t ws_size,
                              hipStream_t stream) {
    const float* feats = (const float*)d_in[0];
    const float* Wfc   = (const float*)d_in[1];
    const float* Wl    = (const float*)d_in[2];
    const float* Wr    = (const float*)d_in[3];
    const float* Wres  = (const float*)d_in[4];
    const int*   esrc  = (const int*)d_in[5];
    const int*   edst  = (const int*)d_in[6];

    int n = in_sizes[0] / IN_DIM;
    int e = in_sizes[5];
    float* out = (float*)d_out;

    // workspace layout
    float* ft = (float*)d_ws;                    // n*HID
    float* sd = ft + (size_t)n * HID;            // n
    float* ss = sd + n;                          // n
    float* wl = ss + n;                          // HID
    float* wr = wl + HID;                        // HID
    int*   rs = (int*)(wr + HID);                // n+1

    gat_wsum<<<1, 64, 0, stream>>>(Wl, Wr, wl, wr);
    gat_rowstart<<<(n + 1 + 255) / 256, 256, 0, stream>>>(edst, rs, n, e);

    int tiles = (n + 15) / 16;
    gat_gemm_wmma<<<(tiles + 7) / 8, 256, 0, stream>>>(feats, Wfc, Wres, ft, out, n);
    gat_scores<<<(n + 7) / 8, 256, 0, stream>>>(ft, wl, wr, sd, ss, n);
    gat_aggregate<<<(n + 7) / 8, 256, 0, stream>>>(ft, sd, ss, esrc, rs, out, n);
}